// JPQ_87170656239702
// MI455X (gfx1250) — compile-verified
//
#include <hip/hip_runtime.h>

typedef __attribute__((ext_vector_type(2))) float v2f;
typedef __attribute__((ext_vector_type(8))) float v8f;

#define MCQ_M 96
#define MCQ_K 256
#define SUBD  8
#define NQ    128
#define NEGK  1024
#define DDIM  768          // MCQ_M * SUBD
#define NSC   (NEGK + 1)   // pos in column 0

// ---------------------------------------------------------------------------
// Kernel 1: ADC lookup table via exact f32 WMMA (V_WMMA_F32_16X16X4_F32).
//   LUT[n][m][c] = sum_{s<8} Q[n][m*8+s] * C[m][c][s]
// Batched GEMM per subspace m: Q_m[128x8] x C_m^T[8x256], K=8 done as two
// chained K=4 WMMAs (no padding, no precision loss, all 32 lanes active).
//
// A layout (32-bit, 16x4): lanes 0-15 hold M=lane with VGPR0=K0, VGPR1=K1;
// lanes 16-31 hold K2/K3. B mirrors with lane = column index. So each lane
// supplies the float pair at k = 2*(lane>>4) (chunk 0) / 4+2*(lane>>4) (chunk 1).
// D layout: VGPR r, lanes 0-15 -> M=r, lanes 16-31 -> M=r+8; N = lane&15.
// ---------------------------------------------------------------------------
__global__ void jpq_lut_wmma(const float* __restrict__ q,
                             const float* __restrict__ cent,
                             float* __restrict__ lut) {
  const int wave = blockIdx.x;            // 0 .. 8*96-1
  const int qt   = wave / MCQ_M;          // query tile (16 rows)
  const int m    = wave % MCQ_M;          // subspace
  const int lane = threadIdx.x;           // 0..31 (wave32)
  const int idx  = lane & 15;             // row (A) / col (B) within tile
  const int hsel = lane >> 4;             // 0: K0/K1, 1: K2/K3

  // A fragments for both K-chunks (row qt*16+idx, k pairs)
  const float* qrow = q + (size_t)(qt * 16 + idx) * DDIM + m * SUBD + 2 * hsel;
  v2f a0; a0[0] = qrow[0]; a0[1] = qrow[1];   // k_base 0
  v2f a1; a1[0] = qrow[4]; a1[1] = qrow[5];   // k_base 4

  const int rbase = hsel * 8;

  #pragma unroll 4
  for (int ct = 0; ct < 16; ++ct) {
    // B fragments: column ct*16+idx (centroid index), same k pairs
    const float* crow = cent + (size_t)(m * MCQ_K + ct * 16 + idx) * SUBD + 2 * hsel;
    v2f b0; b0[0] = crow[0]; b0[1] = crow[1];
    v2f b1; b1[0] = crow[4]; b1[1] = crow[5];

    v8f acc;
    #pragma unroll
    for (int i = 0; i < 8; ++i) acc[i] = 0.0f;
    // (neg_a, A, neg_b, B, c_mod, C, reuse_a, reuse_b); chain K=0..3 then 4..7
    acc = __builtin_amdgcn_wmma_f32_16x16x4_f32(
        false, a0, false, b0, (short)0, acc, false, false);
    acc = __builtin_amdgcn_wmma_f32_16x16x4_f32(
        false, a1, false, b1, (short)0, acc, false, false);

    const int col = ct * 16 + idx;
    #pragma unroll
    for (int r = 0; r < 8; ++r) {
      const int row = qt * 16 + rbase + r;
      lut[((size_t)row * MCQ_M + m) * MCQ_K + col] = acc[r];
    }
  }
}

// ---------------------------------------------------------------------------
// Kernel 2: scores[n][k] = sum_m LUT[n][m][codes[pid][m]],  pid from
// pos_pids (k==0) or neg_pids (k-1). TEMP == 1.0 so no scaling needed.
// One thread per (n,k); codes row loaded as 24x int4 (384 B, 16B-aligned).
// ---------------------------------------------------------------------------
__global__ void jpq_scores(const float* __restrict__ lut,
                           const int* __restrict__ codes,
                           const int* __restrict__ neg_pids,
                           const int* __restrict__ pos_pids,
                           float* __restrict__ scores) {
  const int idx = blockIdx.x * blockDim.x + threadIdx.x;
  if (idx >= NQ * NSC) return;
  const int n = idx / NSC;
  const int k = idx % NSC;
  const int pid = (k == 0) ? pos_pids[n] : neg_pids[n * NEGK + (k - 1)];

  const int4*  c4    = (const int4*)(codes + (size_t)pid * MCQ_M);
  const float* lbase = lut + (size_t)n * MCQ_M * MCQ_K;

  float s = 0.0f;
  #pragma unroll 6
  for (int j = 0; j < MCQ_M / 4; ++j) {
    int4 v = c4[j];
    const float* lb = lbase + (size_t)j * 4 * MCQ_K;
    s += lb[v.x];
    s += lb[MCQ_K + v.y];
    s += lb[2 * MCQ_K + v.z];
    s += lb[3 * MCQ_K + v.w];
  }
  scores[idx] = s;
}

// ---------------------------------------------------------------------------
// Kernel 3: loss = mean_n( logsumexp(scores[n,:]) - scores[n,0] ).
// One block, 128 threads; thread t streams query t's 1025 scores with an
// online (max, sum) logsumexp, then LDS tree-reduce to the mean.
// ---------------------------------------------------------------------------
__global__ void jpq_loss(const float* __restrict__ scores,
                         float* __restrict__ out) {
  const int n = threadIdx.x;  // 0..127
  const float* s = scores + (size_t)n * NSC;

  float mx  = -INFINITY;
  float sum = 0.0f;
  for (int k = 0; k < NSC; ++k) {
    const float v = s[k];
    if (v > mx) {
      sum = sum * __expf(mx - v) + 1.0f;
      mx  = v;
    } else {
      sum += __expf(v - mx);
    }
  }
  const float lse  = mx + __logf(sum);
  const float loss = lse - s[0];

  __shared__ float red[NQ];
  red[n] = loss;
  __syncthreads();
  #pragma unroll
  for (int off = NQ / 2; off > 0; off >>= 1) {
    if (n < off) red[n] += red[n + off];
    __syncthreads();
  }
  if (n == 0) out[0] = red[0] * (1.0f / (float)NQ);
}

// ---------------------------------------------------------------------------
extern "C" void kernel_launch(void* const* d_in, const int* in_sizes, int n_in,
                              void* d_out, int out_size, void* d_ws, size_t ws_size,
                              hipStream_t stream) {
  const float* q        = (const float*)d_in[0];  // [128, 768]
  const float* cent     = (const float*)d_in[1];  // [96, 256, 8]
  const int*   codes    = (const int*)d_in[2];    // [1M, 96]
  const int*   neg_pids = (const int*)d_in[3];    // [128, 1024]
  const int*   pos_pids = (const int*)d_in[4];    // [128]

  // Workspace layout: LUT (128*96*256 f32 = 12.6 MB) then scores (128*1025 f32)
  float* lut    = (float*)d_ws;
  float* scores = lut + (size_t)NQ * MCQ_M * MCQ_K;

  // 1) LUT build: 8 query tiles x 96 subspaces, one wave (32 threads) each
  jpq_lut_wmma<<<(NQ / 16) * MCQ_M, 32, 0, stream>>>(q, cent, lut);

  // 2) scores: one thread per (query, slot)
  const int tot = NQ * NSC;
  jpq_scores<<<(tot + 255) / 256, 256, 0, stream>>>(lut, codes, neg_pids, pos_pids, scores);

  // 3) logsumexp + CE mean -> scalar
  jpq_loss<<<1, NQ, 0, stream>>>(scores, (float*)d_out);
}